// LinearReg_55508157333593
// MI455X (gfx1250) — compile-verified
//
#include <hip/hip_runtime.h>
#include <hip/hip_bf16.h>
#include <stdint.h>

// ---- problem constants (reference: [N, 32*25] fp32, group L2 norms) ----
#define GROUPS     32
#define GSIZE      25
#define ROWLEN     800                  // floats per row
#define TILE_ROWS  8                    // one row per wave at compute time
#define TILE_F     (TILE_ROWS * ROWLEN) // 6400 floats per tile
#define NTHREADS   256                  // 8 waves of 32 (wave32)
#define NBLOCKS    1024

#if defined(__gfx1250__) && __has_builtin(__builtin_amdgcn_global_load_async_to_lds_b128)
#define USE_ASYNC 1
#else
#define USE_ASYNC 0
#endif

typedef int v4i __attribute__((vector_size(16)));
typedef __attribute__((address_space(1))) v4i GlobalV4i;  // global (AS1)
typedef __attribute__((address_space(3))) v4i LdsV4i;     // LDS (AS3)

__device__ __forceinline__ void copy_tile(float* __restrict__ dst_lds,
                                          const float* __restrict__ src,
                                          int n4, int tid) {
  const float4* __restrict__ s4 = (const float4*)src;
  float4* __restrict__ d4 = (float4*)dst_lds;
#if USE_ASYNC
  for (int i = tid; i < n4; i += NTHREADS) {
    __builtin_amdgcn_global_load_async_to_lds_b128(
        (GlobalV4i*)(s4 + i),
        (LdsV4i*)(d4 + i),
        /*offset=*/0, /*cpol=*/0);
  }
#else
  for (int i = tid; i < n4; i += NTHREADS) {
    d4[i] = s4[i];
  }
#endif
}

__device__ __forceinline__ void wait_async_done() {
#if USE_ASYNC
#if __has_builtin(__builtin_amdgcn_s_wait_asynccnt)
  __builtin_amdgcn_s_wait_asynccnt(0);
#else
  asm volatile("s_wait_asynccnt 0x0" ::: "memory");
#endif
#endif
}

__global__ __launch_bounds__(NTHREADS)
void group_norm_partial_kernel(const float* __restrict__ w,
                               float* __restrict__ partials,
                               int rows) {
  __shared__ alignas(16) float buf[2][TILE_F];  // 51.2 KB double buffer
  __shared__ float wave_sums[NTHREADS / 32];

  const int tid  = threadIdx.x;
  const int wave = tid >> 5;
  const int lane = tid & 31;
  const int ntiles = (rows + TILE_ROWS - 1) / TILE_ROWS;

  float acc = 0.0f;
  int cur = 0;
  int t = blockIdx.x;

  // prologue: fill buffer 0
  if (t < ntiles) {
    int tr = rows - t * TILE_ROWS; if (tr > TILE_ROWS) tr = TILE_ROWS;
    copy_tile(buf[0], w + (size_t)t * TILE_F, tr * (ROWLEN / 4), tid);
  }
  wait_async_done();
  __syncthreads();

  for (; t < ntiles; t += gridDim.x) {
    // issue async fill of the other buffer for this block's next tile
    int nt = t + gridDim.x;
    if (nt < ntiles) {
      int tr = rows - nt * TILE_ROWS; if (tr > TILE_ROWS) tr = TILE_ROWS;
      copy_tile(buf[cur ^ 1], w + (size_t)nt * TILE_F, tr * (ROWLEN / 4), tid);
    }

    // compute on current buffer: wave -> row, lane -> 25-element group
    int row = t * TILE_ROWS + wave;
    if (row < rows) {
      const float* g = &buf[cur][wave * ROWLEN + lane * GSIZE];
      float s = 0.0f;
#pragma unroll
      for (int i = 0; i < GSIZE; ++i) {
        float x = g[i];
        s = fmaf(x, x, s);               // bank-conflict-free: gcd(25,64)==1
      }
      float nrm = sqrtf(s);
#pragma unroll
      for (int off = 16; off > 0; off >>= 1)
        nrm += __shfl_xor(nrm, off, 32); // wave32 butterfly, fixed order
      if (lane == 0) acc += nrm;
    }

    __syncthreads();     // all reads of buf[cur] complete before it is reused
    wait_async_done();   // this wave's async fills of buf[cur^1] landed
    __syncthreads();     // every wave's fills landed
    cur ^= 1;
  }

  if (lane == 0) wave_sums[wave] = acc;
  __syncthreads();
  if (tid == 0) {
    float b = 0.0f;
#pragma unroll
    for (int i = 0; i < NTHREADS / 32; ++i) b += wave_sums[i];  // fixed order
    partials[blockIdx.x] = b;
  }
}

__global__ __launch_bounds__(NTHREADS)
void final_reduce_kernel(const float* __restrict__ partials, int nb,
                         const int* __restrict__ c_omega,
                         float* __restrict__ out, int rows) {
  __shared__ float s[NTHREADS];
  float a = 0.0f;
  for (int i = threadIdx.x; i < nb; i += NTHREADS) a += partials[i]; // fixed order
  s[threadIdx.x] = a;
  __syncthreads();
  for (int off = NTHREADS / 2; off > 0; off >>= 1) {
    if (threadIdx.x < off) s[threadIdx.x] += s[threadIdx.x + off];
    __syncthreads();
  }
  if (threadIdx.x == 0) {
    out[0] = (s[0] / (float)rows) * (0.001f * (float)c_omega[0]);
  }
}

extern "C" void kernel_launch(void* const* d_in, const int* in_sizes, int n_in,
                              void* d_out, int out_size, void* d_ws, size_t ws_size,
                              hipStream_t stream) {
  const float* weight   = (const float*)d_in[0];
  const int*   c_omega  = (const int*)d_in[1];
  float*       out      = (float*)d_out;
  float*       partials = (float*)d_ws;          // NBLOCKS floats (4 KB)

  const int rows = in_sizes[0] / (GROUPS * GSIZE);

  group_norm_partial_kernel<<<dim3(NBLOCKS), dim3(NTHREADS), 0, stream>>>(
      weight, partials, rows);
  final_reduce_kernel<<<dim3(1), dim3(NTHREADS), 0, stream>>>(
      partials, NBLOCKS, c_omega, out, rows);
}